// MovingAverageThreshold_1451698946367
// MI455X (gfx1250) — compile-verified
//
#include <hip/hip_runtime.h>
#include <cmath>

typedef float v2f __attribute__((ext_vector_type(2)));
typedef float v8f __attribute__((ext_vector_type(8)));
typedef float f4  __attribute__((ext_vector_type(4)));

#define WG 1024

// ---------------- wave helpers (wave32) ----------------
__device__ __forceinline__ float wave_incl_scan(float x, int lane) {
#pragma unroll
  for (int d = 1; d < 32; d <<= 1) {
    float t = __shfl_up(x, d, 32);
    if (lane >= d) x += t;
  }
  return x;
}
__device__ __forceinline__ float wave_reduce_sum(float x) {
#pragma unroll
  for (int d = 16; d > 0; d >>= 1) x += __shfl_down(x, d, 32);
  return x;
}
__device__ __forceinline__ float wave_reduce_min(float x) {
#pragma unroll
  for (int d = 16; d > 0; d >>= 1) x = fminf(x, __shfl_down(x, d, 32));
  return x;
}

// ---------------- kernel 1: zero histogram ----------------
__global__ void k_zero(float* __restrict__ hist, int n) {
  int i = blockIdx.x * blockDim.x + threadIdx.x;
  if (i < n) hist[i] = 0.0f;
}

// ---------------- kernel 2: scatter-add histogram (vec4) ----------------
__global__ void k_hist(const f4* __restrict__ stat, const f4* __restrict__ dyn,
                       const f4* __restrict__ score, float* __restrict__ hist,
                       int n4, int res) {
  int i = blockIdx.x * blockDim.x + threadIdx.x;
  if (i >= n4) return;
  f4 s  = __builtin_nontemporal_load(&stat[i]);
  f4 d  = __builtin_nontemporal_load(&dyn[i]);
  f4 sc = __builtin_nontemporal_load(&score[i]);
#pragma unroll
  for (int j = 0; j < 4; ++j) {
    float imp = s[j] - d[j];
    int idx = (int)((float)res * sc[j]);   // truncating cast like astype(int32)
    idx = idx > res - 1 ? res - 1 : idx;
    idx = idx < 0 ? 0 : idx;
    unsafeAtomicAdd(&hist[idx], imp);      // native global_atomic_add_f32
  }
}

__global__ void k_hist_tail(const float* __restrict__ stat, const float* __restrict__ dyn,
                            const float* __restrict__ score, float* __restrict__ hist,
                            int start, int n, int res) {
  int i = start + blockIdx.x * blockDim.x + threadIdx.x;
  if (i >= n) return;
  float imp = stat[i] - dyn[i];
  int idx = (int)((float)res * score[i]);
  idx = idx > res - 1 ? res - 1 : idx;
  idx = idx < 0 ? 0 : idx;
  unsafeAtomicAdd(&hist[idx], imp);
}

// ---------------- kernel 3: per-block sums of mai ----------------
__global__ void k_bsum(const float* __restrict__ hist, const float* __restrict__ ma,
                       float w, float* __restrict__ bsum, int n) {
  __shared__ float s[32];
  int tid = threadIdx.x, lane = tid & 31, wid = tid >> 5;
  int i = blockIdx.x * WG + tid;
  float v = 0.0f;
  if (i < n) v = ma[i] * w + (1.0f - w) * hist[i];
  v = wave_reduce_sum(v);
  if (lane == 0) s[wid] = v;
  __syncthreads();
  if (wid == 0) {
    float t = s[lane];
    t = wave_reduce_sum(t);
    if (lane == 0) bsum[blockIdx.x] = t;
  }
}

// ---------------- kernel 4: exclusive scan of block sums (tiny) ----------------
__global__ void k_boff(const float* __restrict__ bsum, float* __restrict__ boff,
                       float* __restrict__ prefix, int nb) {
  __shared__ float s[256];
  int tid = threadIdx.x;
  if (tid < nb) s[tid] = bsum[tid];
  __syncthreads();
  if (tid == 0) {
    float acc = 0.0f;
    for (int b = 0; b < nb; ++b) { boff[b] = acc; acc += s[b]; }
    prefix[0] = 0.0f;   // improv_over_thresh[0]
  }
}

// ---------------- kernel 5: block scans -> prefix array ----------------
__global__ void k_scan(const float* __restrict__ hist, const float* __restrict__ ma,
                       const float* __restrict__ boff, float w,
                       float* __restrict__ prefix, int n) {
  __shared__ float s[32];
  int tid = threadIdx.x, lane = tid & 31, wid = tid >> 5;
  int i = blockIdx.x * WG + tid;
  float v = (i < n) ? ma[i] * w + (1.0f - w) * hist[i] : 0.0f;
  float x = wave_incl_scan(v, lane);
  if (lane == 31) s[wid] = x;
  __syncthreads();
  if (wid == 0) {
    float t = s[lane];
    t = wave_incl_scan(t, lane);
    s[lane] = t;                 // inclusive sums of wave totals
  }
  __syncthreads();
  float offs = (wid > 0) ? s[wid - 1] : 0.0f;
  if (i < n) prefix[i + 1] = x + offs + boff[blockIdx.x];
}

// ---------------- kernel 6: min + WMMA tie dot-products + final ----------------
__global__ void k_final(const float* __restrict__ prefix,
                        const float* __restrict__ bias_counter,
                        float w, float* __restrict__ out, int np, int res) {
  __shared__ float sred[32];
  __shared__ float sbest, scnt, sidx;
  int tid = threadIdx.x, lane = tid & 31, wid = tid >> 5;

  // --- global min over prefix[0..np-1] ---
  float m = 3.402823466e38f;
  for (int i = tid; i < np; i += WG) m = fminf(m, prefix[i]);
  m = wave_reduce_min(m);
  if (lane == 0) sred[wid] = m;
  __syncthreads();
  if (tid == 0) {
    float b = sred[0];
    for (int k = 1; k < 32; ++k) b = fminf(b, sred[k]);
    sbest = b; scnt = 0.0f; sidx = 0.0f;
  }
  __syncthreads();
  float best = sbest;

  // --- tie count & tie index-sum as accumulating WMMA dot products ---
  // A-matrix (16x4 f32, documented layout): lanes 0-15 hold row M=lane at K=0,1;
  // lanes 16-31 hold row M=lane-16 at K=2,3. Element e = base + 4*M + K.
  // B = all-ones (layout-independent) => D[m][n] = sum_k A[m][k] + C[m][n].
  int mrow  = lane & 15;
  int kbase = (lane >> 4) * 2;
  v2f bones; bones[0] = 1.0f; bones[1] = 1.0f;
  v8f cc = {};   // accumulates count row-sums
  v8f ci = {};   // accumulates index-weighted row-sums
  for (int base = wid * 64; base < np; base += 32 * 64) {
    int i0 = base + 4 * mrow + kbase;
    int i1 = i0 + 1;
    int c0 = i0 < np ? i0 : np - 1;
    int c1 = i1 < np ? i1 : np - 1;
    float p0 = prefix[c0];
    float p1 = prefix[c1];
    float m0 = (i0 < np && p0 == best) ? 1.0f : 0.0f;
    float m1 = (i1 < np && p1 == best) ? 1.0f : 0.0f;
    v2f a;  a[0]  = m0;              a[1]  = m1;
    v2f ai; ai[0] = m0 * (float)i0;  ai[1] = m1 * (float)i1;
    cc = __builtin_amdgcn_wmma_f32_16x16x4_f32(false, a,  false, bones,
                                               (short)0, cc, false, false);
    ci = __builtin_amdgcn_wmma_f32_16x16x4_f32(false, ai, false, bones,
                                               (short)0, ci, false, false);
  }
  // C/D layout: reg r, lane L -> row M = r + 8*(L>=16), col N = L%16 (cols duplicated).
  float pc = 0.0f, pi = 0.0f;
#pragma unroll
  for (int r = 0; r < 8; ++r) { pc += cc[r]; pi += ci[r]; }
  pc += __shfl_down(pc, 16, 32);   // rows 0-7 + rows 8-15
  pi += __shfl_down(pi, 16, 32);
  if (lane == 0) {
    atomicAdd(&scnt, pc);          // ds_add_f32
    atomicAdd(&sidx, pi);
  }
  __syncthreads();
  if (tid == 0) {
    float avg  = sidx / scnt;
    float thr  = avg * (1.0f / (float)res);     // VALUE_LO + avg*SPAN/RES
    float bias = bias_counter[0] * w + (1.0f - w);
    out[0] = (bias > 0.0f) ? thr : 0.5f;
  }
}

extern "C" void kernel_launch(void* const* d_in, const int* in_sizes, int n_in,
                              void* d_out, int out_size, void* d_ws, size_t ws_size,
                              hipStream_t stream) {
  const float* stat = (const float*)d_in[0];
  const float* dyn  = (const float*)d_in[1];
  const float* sc   = (const float*)d_in[2];
  const float* ma   = (const float*)d_in[3];
  const float* bias = (const float*)d_in[4];
  float* out = (float*)d_out;

  int n   = in_sizes[0];       // 8,000,000
  int res = in_sizes[3];       // 100,000

  float* ws     = (float*)d_ws;
  float* hist   = ws;                      // res floats
  float* prefix = hist + res;              // res+1 floats
  float* bsum   = prefix + (res + 1);      // <=256 floats
  float* boff   = bsum + 256;              // <=256 floats

  // (1 - uw)^n in double (1-2e-9 rounds to 1.0 in f32)
  double uw = 1.0 / fmin(2.0 * 2000000000.0, 5000.0 * (2000000000.0 / 20000.0));
  float  w  = (float)std::pow(1.0 - uw, (double)n);

  k_zero<<<(res + 255) / 256, 256, 0, stream>>>(hist, res);

  int n4 = n / 4;
  if (n4 > 0)
    k_hist<<<(n4 + 255) / 256, 256, 0, stream>>>(
        (const f4*)stat, (const f4*)dyn, (const f4*)sc, hist, n4, res);
  int tail = n - n4 * 4;
  if (tail > 0)
    k_hist_tail<<<1, 256, 0, stream>>>(stat, dyn, sc, hist, n4 * 4, n, res);

  int nb = (res + WG - 1) / WG;            // 98 blocks
  k_bsum<<<nb, WG, 0, stream>>>(hist, ma, w, bsum, res);
  k_boff<<<1, 256, 0, stream>>>(bsum, boff, prefix, nb);
  k_scan<<<nb, WG, 0, stream>>>(hist, ma, boff, w, prefix, res);
  k_final<<<1, WG, 0, stream>>>(prefix, bias, w, out, res + 1, res);
}